// RealmReaderProjection_55336358643252
// MI455X (gfx1250) — compile-verified
//
#include <hip/hip_runtime.h>
#include <hip/hip_bf16.h>

typedef float v2f __attribute__((ext_vector_type(2)));
typedef float v4f __attribute__((ext_vector_type(4)));
typedef float v8f __attribute__((ext_vector_type(8)));

#define B_    64
#define L_    512
#define H_    768      // K of the GEMM
#define D_    256
#define N_    512      // 2*D
#define S_    5075     // sum_{w=0..9} (512 - w)
#define M_    (B_ * L_)   // 32768
#define EPS_  1e-3f
#define FMIN_ (-3.4028234663852886e+38f)

#define KC_     32               // K chunk staged in LDS
#define LDSTR_  36               // padded row stride (floats): 32->36 kills bank conflicts
#define LDSA_N  (32 * LDSTR_)    // A chunk: 32 rows
#define LDSB_N  (512 * LDSTR_)   // B chunk: 512 rows

// gfx1250 async global->LDS copy (ASYNCcnt path); 16B per lane per issue.
__device__ __forceinline__ void async_ld_b128(unsigned lds_off, const float* g) {
    asm volatile("global_load_async_to_lds_b128 %0, %1, off"
                 :: "v"(lds_off), "v"((unsigned long long)(size_t)g)
                 : "memory");
}
__device__ __forceinline__ void wait_async0() {
    asm volatile("s_wait_asynccnt 0x0" ::: "memory");
}

// ---------------------------------------------------------------------------
// Kernel 1: h[M,512] = HS[M,768] @ W_int[512,768]^T + b_int    (fp32 WMMA)
//
// Block = 256 threads (8 waves) computes a 32(M) x 512(N) tile.
// Per K-chunk of 32: async-copy A(32x32)+B(512x32) into LDS (coalesced 16B
// segments, ASYNCcnt), then each wave runs 8 x V_WMMA_F32_16X16X4_F32 per
// 4-wide K step from conflict-free ds_load_b64 fragments.
//
// V_WMMA_F32_16X16X4_F32 operand layouts (ISA 7.12.2, wave32):
//   A 16x4 : lanes 0-15 = M rows; VGPR pair = K{0,1} (lanes 0-15) / K{2,3}
//   B 4x16 : mirrored;  C/D: VGPR v -> (M=v | v+8, N=lane&15)
// ---------------------------------------------------------------------------
__global__ __launch_bounds__(256)
void rrp_gemm_wmma(const float* __restrict__ HS, const float* __restrict__ W,
                   const float* __restrict__ bias, float* __restrict__ Hout) {
    __shared__ float ldsA[LDSA_N];
    __shared__ float ldsB[LDSB_N];

    const int tid  = threadIdx.x;
    const int lane = tid & 31;
    const int warp = tid >> 5;
    const int mBlk = blockIdx.x * 32;        // 1024 blocks exactly cover M
    const int mSub = (warp & 1) * 16;        // wave's M sub-tile inside block
    const int nT   = (warp >> 1) * 128;      // wave's N tile (0/128/256/384)
    const int lm   = lane & 15;
    const int hf   = lane >> 4;

    // async-copy geometry: 16B segments, 8 segments per 32-float row
    const int rowA = tid >> 3;               // 0..31
    const int segA = tid & 7;                // 0..7

    v8f acc[8] = {};

    for (int kc = 0; kc < H_; kc += KC_) {
        __syncthreads();                     // previous chunk fully consumed

        // A chunk: 32 rows x 32 cols, one 16B segment per thread
        async_ld_b128((unsigned)(size_t)&ldsA[rowA * LDSTR_ + segA * 4],
                      HS + (size_t)(mBlk + rowA) * H_ + kc + segA * 4);
        // B chunk: 512 rows x 32 cols, 16 segments per thread
#pragma unroll
        for (int i = 0; i < 16; ++i) {
            const int t = tid + i * 256;     // 0..4095
            const int n = t >> 3, seg = t & 7;
            async_ld_b128((unsigned)(size_t)&ldsB[n * LDSTR_ + seg * 4],
                          W + (size_t)n * H_ + kc + seg * 4);
        }
        wait_async0();                       // own async copies landed in LDS
        __syncthreads();                     // everyone's copies visible

#pragma unroll
        for (int kk = 0; kk < KC_; kk += 4) {
            v2f a = *(const v2f*)&ldsA[(mSub + lm) * LDSTR_ + kk + 2 * hf];
#pragma unroll
            for (int j = 0; j < 8; ++j) {
                v2f b = *(const v2f*)&ldsB[(nT + j * 16 + lm) * LDSTR_ + kk + 2 * hf];
                // (neg_a, A, neg_b, B, c_mod, C, reuse_a, reuse_b)
                acc[j] = __builtin_amdgcn_wmma_f32_16x16x4_f32(
                    false, a, false, b, (short)0, acc[j], false, false);
            }
        }
    }

#pragma unroll
    for (int j = 0; j < 8; ++j) {
        const int   n  = nT + j * 16 + lm;
        const float bn = bias[n];
#pragma unroll
        for (int v = 0; v < 8; ++v) {
            const int m = mBlk + mSub + v + 8 * hf;
            Hout[(size_t)m * N_ + n] = acc[j][v] + bn;
        }
    }
}

// ---------------------------------------------------------------------------
// Kernel 2: one wave32 per span.
//   cand = relu(start_proj[b,start] + end_proj[b,end])          (D=256)
//   layernorm over D, then rank-1 projection + bias + block mask
// Each lane owns 8 elements as two float4 chunks -> 512B coalesced loads.
// Cross-lane reductions: 5-step __shfl_xor (wave32).
// ---------------------------------------------------------------------------
__global__ __launch_bounds__(256, 4)
void rrp_span(const float* __restrict__ Hp, const unsigned char* __restrict__ mask,
              const float* __restrict__ gamma, const float* __restrict__ beta,
              const float* __restrict__ wout, const float* __restrict__ bout,
              float* __restrict__ logits) {
    const int lane = threadIdx.x & 31;
    const int warp = threadIdx.x >> 5;
    const int wid  = blockIdx.x * 8 + warp;
    if (wid >= B_ * S_) return;                  // uniform per wave
    const int b = wid / S_;
    const int s = wid - b * S_;

    // decode flat span index -> (start, end); widths are grouped: w = end-start
    int w = 0, off = 0;
    while (s >= off + (L_ - w)) { off += (L_ - w); ++w; }
    const int start = s - off;
    const int end   = start + w;

    const float* sp = Hp + (size_t)(b * L_ + start) * N_;        // first D cols
    const float* ep = Hp + (size_t)(b * L_ + end)   * N_ + D_;   // second D cols

    v4f c0, c1;
    {
        v4f s0 = *(const v4f*)(sp + lane * 4);
        v4f s1 = *(const v4f*)(sp + 128 + lane * 4);
        v4f e0 = *(const v4f*)(ep + lane * 4);
        v4f e1 = *(const v4f*)(ep + 128 + lane * 4);
        c0 = s0 + e0;
        c1 = s1 + e1;
    }
#pragma unroll
    for (int i = 0; i < 4; ++i) {
        c0[i] = fmaxf(c0[i], 0.f);
        c1[i] = fmaxf(c1[i], 0.f);
    }

    // mean
    float sum = c0[0] + c0[1] + c0[2] + c0[3] + c1[0] + c1[1] + c1[2] + c1[3];
#pragma unroll
    for (int o = 16; o > 0; o >>= 1) sum += __shfl_xor(sum, o, 32);
    const float mu = sum * (1.f / D_);

    // variance (two-pass, data already in registers)
    float vs = 0.f;
#pragma unroll
    for (int i = 0; i < 4; ++i) {
        float d0 = c0[i] - mu; vs += d0 * d0;
        float d1 = c1[i] - mu; vs += d1 * d1;
    }
#pragma unroll
    for (int o = 16; o > 0; o >>= 1) vs += __shfl_xor(vs, o, 32);
    const float rstd = rsqrtf(vs * (1.f / D_) + EPS_);

    // rank-1 projection of normed values
    v4f g0 = *(const v4f*)(gamma + lane * 4), g1 = *(const v4f*)(gamma + 128 + lane * 4);
    v4f t0 = *(const v4f*)(beta  + lane * 4), t1 = *(const v4f*)(beta  + 128 + lane * 4);
    v4f o0 = *(const v4f*)(wout  + lane * 4), o1 = *(const v4f*)(wout  + 128 + lane * 4);
    float dot = 0.f;
#pragma unroll
    for (int i = 0; i < 4; ++i) {
        dot += ((c0[i] - mu) * rstd * g0[i] + t0[i]) * o0[i];
        dot += ((c1[i] - mu) * rstd * g1[i] + t1[i]) * o1[i];
    }
#pragma unroll
    for (int o = 16; o > 0; o >>= 1) dot += __shfl_xor(dot, o, 32);

    if (lane == 0) {
        float logit = dot + bout[0];
        const bool ok = mask[b * L_ + start] && mask[b * L_ + end];
        if (!ok) logit += FMIN_;                 // matches "+ (1-m)*finfo.min"
        logits[(size_t)b * S_ + s] = logit;
    }
}

// ---------------------------------------------------------------------------
// Kernel 3: starts[S] and ends[S] int32 tables appended after logits.
// ---------------------------------------------------------------------------
__global__ void rrp_indices(int* __restrict__ idxOut) {
    const int t = blockIdx.x * 256 + threadIdx.x;
    if (t >= S_) return;
    int w = 0, off = 0;
    while (t >= off + (L_ - w)) { off += (L_ - w); ++w; }
    idxOut[t]      = t - off;        // start
    idxOut[S_ + t] = t - off + w;    // end
}

extern "C" void kernel_launch(void* const* d_in, const int* in_sizes, int n_in,
                              void* d_out, int out_size, void* d_ws, size_t ws_size,
                              hipStream_t stream) {
    (void)in_sizes; (void)n_in; (void)out_size; (void)ws_size;

    const float*         HS    = (const float*)d_in[0];         // [64,512,768]
    const unsigned char* mask  = (const unsigned char*)d_in[1]; // [64,512] bool
    const float*         Wint  = (const float*)d_in[2];         // [512,768]
    const float*         bint  = (const float*)d_in[3];         // [512]
    const float*         gamma = (const float*)d_in[4];         // [256]
    const float*         beta  = (const float*)d_in[5];         // [256]
    const float*         wout  = (const float*)d_in[6];         // [256]
    const float*         bout  = (const float*)d_in[7];         // [1]

    float* logits = (float*)d_out;                              // [64*5075]
    int*   idxOut = (int*)((float*)d_out + (size_t)B_ * S_);    // starts, ends
    float* Hproj  = (float*)d_ws;                               // 64 MB scratch

    // GEMM: 1024 blocks x (8 waves, 32x512 tile) -- exact cover, EXEC all-ones
    rrp_gemm_wmma<<<M_ / 32, 256, 0, stream>>>(HS, Wint, bint, Hproj);

    const int totalWaves = B_ * S_;                             // 324800
    rrp_span<<<(totalWaves + 7) / 8, 256, 0, stream>>>(Hproj, mask, gamma, beta,
                                                       wout, bout, logits);

    rrp_indices<<<(S_ + 255) / 256, 256, 0, stream>>>(idxOut);
}